// Pool_25503515803828
// MI455X (gfx1250) — compile-verified
//
#include <hip/hip_runtime.h>
#include <hip/hip_bf16.h>
#include <stdint.h>

// Segment-mean pool + gather-broadcast, MI455X (gfx1250, wave32).
// d = 64 floats per row (16 float4 lanes per row).
#define DFEAT  64
#define DQ     16          // float4s per row
#define EPS_F  1e-16f

typedef float v4f __attribute__((ext_vector_type(4)));

// ---- gfx1250 async global->LDS load (ASYNCcnt path), inline-asm form -------
// Per ISA §10.2 the low 32 bits of a generic (flat) LDS pointer are the LDS
// byte offset, so we can feed it straight into the VDST (LDS address) VGPR.
__device__ __forceinline__ void async_ld_b128(const void* gaddr, void* lds_generic) {
    unsigned lofs = (unsigned)(uintptr_t)lds_generic;
    asm volatile("global_load_async_to_lds_b128 %0, %1, off"
                 :: "v"(lofs), "v"((unsigned long long)gaddr)
                 : "memory");
}

template <int N>
__device__ __forceinline__ void wait_async() {
#if __has_builtin(__builtin_amdgcn_s_wait_asynccnt)
    __builtin_amdgcn_s_wait_asynccnt(N);
#else
    asm volatile("s_wait_asynccnt %0" :: "n"(N));
#endif
    asm volatile("" ::: "memory");   // keep LDS reads below the wait
}

// ---- fp32 HW atomic add (global_atomic_add_f32, no-return form) ------------
__device__ __forceinline__ void atomic_add_f32(float* p, float v) {
    unsafeAtomicAdd(p, v);
}

// ws layout (floats): [0, C*64) sums | [C*64, C*65) cnt | [C*65, C*66) inv
// ---------------------------------------------------------------------------

__global__ void pool_zero_ws(float* __restrict__ ws, const int* __restrict__ ncols_p) {
    const int C = *ncols_p;
    const size_t n = (size_t)C * (DFEAT + 1);   // sums + cnt (inv fully overwritten)
    const size_t stride = (size_t)gridDim.x * blockDim.x;
    for (size_t i = (size_t)blockIdx.x * blockDim.x + threadIdx.x; i < n; i += stride)
        ws[i] = 0.0f;
}

__global__ void pool_scatter(const float* __restrict__ vals,
                             const long long* __restrict__ seg,
                             float* __restrict__ ws,
                             const int* __restrict__ ncols_p,
                             int nnz) {
    const int C = *ncols_p;
    float* __restrict__ sums = ws;
    float* __restrict__ cnt  = ws + (size_t)C * DFEAT;
    const size_t total  = (size_t)nnz * DQ;
    const size_t stride = (size_t)gridDim.x * blockDim.x;
    for (size_t i = (size_t)blockIdx.x * blockDim.x + threadIdx.x; i < total; i += stride) {
        const size_t row = i >> 4;
        const int    q   = (int)(i & 15);
        const int    col = (int)seg[row];
        const v4f v = __builtin_nontemporal_load(
            (const v4f*)(vals + (row << 6) + ((size_t)q << 2)));
        float* dst = sums + ((size_t)col << 6) + ((size_t)q << 2);
        atomic_add_f32(dst + 0, v.x);
        atomic_add_f32(dst + 1, v.y);
        atomic_add_f32(dst + 2, v.z);
        atomic_add_f32(dst + 3, v.w);
        if (q == 0) atomic_add_f32(cnt + col, 1.0f);
    }
}

__global__ void pool_invcnt(float* __restrict__ ws, const int* __restrict__ ncols_p) {
    const int C = *ncols_p;
    const float* __restrict__ cnt = ws + (size_t)C * DFEAT;
    float* __restrict__ inv = ws + (size_t)C * (DFEAT + 1);
    const int stride = gridDim.x * blockDim.x;
    for (int c = blockIdx.x * blockDim.x + threadIdx.x; c < C; c += stride)
        inv[c] = 1.0f / (cnt[c] + EPS_F);
}

// Gather: per 256-thread block, a tile = 16 rows; thread t handles float4 #(t&15)
// of row (t>>4). Double-buffered async LDS staging of mean rows from L2.
__global__ void pool_gather(const float* __restrict__ ws,
                            const int* __restrict__ ncols_p,
                            const long long* __restrict__ seg,
                            float* __restrict__ out,
                            int nnz) {
    const int C = *ncols_p;
    const float* __restrict__ sums = ws;
    const float* __restrict__ inv  = ws + (size_t)C * (DFEAT + 1);

    __shared__ __align__(16) float lds[2][256 * 4];   // 2 x 4 KB

    const int tid   = threadIdx.x;
    const int r     = tid >> 4;      // row within tile
    const int q     = tid & 15;      // float4 within row
    const int tiles = (nnz + 15) >> 4;
    const int gstr  = gridDim.x;

    int tile = blockIdx.x;
    if (tile >= tiles) return;       // uniform per block

    auto issue = [&](int t, int buf) -> int {
        int row = t * 16 + r;
        if (row >= nnz) row = nnz - 1;          // clamp: EXEC stays all-ones
        const int col = (int)seg[row];
        const float* g = sums + ((size_t)col << 6) + ((size_t)q << 2);
        async_ld_b128(g, (void*)&lds[buf][(size_t)tid << 2]);
        return col;
    };

    int buf = 0;
    int colCur = issue(tile, buf);               // prime the pipeline
    while (true) {
        const int nt = tile + gstr;              // uniform per block
        int colNext = 0;
        if (nt < tiles) {
            colNext = issue(nt, buf ^ 1);        // prefetch next tile
            wait_async<1>();                     // oldest (current tile) done
        } else {
            wait_async<0>();
        }
        const v4f m = *(const v4f*)&lds[buf][(size_t)tid << 2];
        const float s = inv[colCur];
        const int row  = tile * 16 + r;
        if (row < nnz) {
            v4f o;
            o.x = m.x * s; o.y = m.y * s; o.z = m.z * s; o.w = m.w * s;
            __builtin_nontemporal_store(
                o, (v4f*)(out + ((size_t)row << 6) + ((size_t)q << 2)));
        }
        if (nt >= tiles) break;
        tile = nt; colCur = colNext; buf ^= 1;
    }
}

extern "C" void kernel_launch(void* const* d_in, const int* in_sizes, int n_in,
                              void* d_out, int out_size, void* d_ws, size_t ws_size,
                              hipStream_t stream) {
    (void)n_in; (void)out_size; (void)ws_size;
    const float*     vals  = (const float*)d_in[0];
    const long long* idx   = (const long long*)d_in[1];   // [2, nnz] int64
    const int*       ncols = (const int*)d_in[2];         // LE: low 32b ok for i32/i64
    const int nnz = in_sizes[1] / 2;
    if (nnz <= 0) return;
    const long long* seg = idx + nnz;                     // indices[1] = column ids
    float* ws  = (float*)d_ws;
    float* out = (float*)d_out;

    // 1) zero sums + counts (ws is poisoned before timing)
    pool_zero_ws<<<2048, 256, 0, stream>>>(ws, ncols);

    // 2) scatter-add into L2-resident accumulators
    const size_t scat_work = (size_t)nnz * DQ;
    int scat_blocks = (int)((scat_work + 255) / 256);
    if (scat_blocks > 131072) scat_blocks = 131072;
    pool_scatter<<<scat_blocks, 256, 0, stream>>>(vals, seg, ws, ncols, nnz);

    // 3) per-segment reciprocal counts
    pool_invcnt<<<256, 256, 0, stream>>>(ws, ncols);

    // 4) async-LDS double-buffered gather + scale + NT store
    const int tiles = (nnz + 15) / 16;
    int gb = tiles < 4096 ? tiles : 4096;
    pool_gather<<<gb, 256, 0, stream>>>(ws, ncols, seg, out, nnz);
}